// SAGEConv_89799176224850
// MI455X (gfx1250) — compile-verified
//
#include <hip/hip_runtime.h>

#define N_NODES 50000
#define N_EDGES 800000
#define D 64

typedef __attribute__((ext_vector_type(2))) float v2f;
typedef __attribute__((ext_vector_type(8))) float v8f;

// ---------------------------------------------------------------------------
// Kernel 0: zero neigh_sum + deg workspace (harness poisons d_ws with 0xAA).
// ---------------------------------------------------------------------------
__global__ void zero_ws_kernel(float* __restrict__ ws, int n) {
    int i = blockIdx.x * blockDim.x + threadIdx.x;
    int stride = gridDim.x * blockDim.x;
    for (; i < n; i += stride) ws[i] = 0.0f;
}

// ---------------------------------------------------------------------------
// Kernel 1: edge scatter-add. One wave32 per edge; lane L handles feature
// dims L and L+32 -> coalesced 128B gather + coalesced f32 atomics into L2.
// Lane 0 accumulates the in-degree.
// ---------------------------------------------------------------------------
__global__ __launch_bounds__(256) void edge_scatter_kernel(
    const float* __restrict__ feat,
    const int*   __restrict__ src,
    const int*   __restrict__ dst,
    float*       __restrict__ neigh_sum,
    float*       __restrict__ deg) {
    int wave = (blockIdx.x * blockDim.x + threadIdx.x) >> 5;
    int lane = threadIdx.x & 31;
    if (wave >= N_EDGES) return;
    int s = src[wave];
    int d = dst[wave];
    float f0 = feat[(size_t)s * D + lane];
    float f1 = feat[(size_t)s * D + lane + 32];
    atomicAdd(&neigh_sum[(size_t)d * D + lane],      f0);
    atomicAdd(&neigh_sum[(size_t)d * D + lane + 32], f1);
    if (lane == 0) atomicAdd(&deg[d], 1.0f);
}

// ---------------------------------------------------------------------------
// Kernel 2: out = ReLU(feat @ Wself^T + bself + (neigh_sum/deg) @ Wneigh^T + bneigh)
// 256 threads = 8 waves; each wave owns one 16x16 output tile of a 32x64
// block tile. K-loop: 16 steps of V_WMMA_F32_16X16X4_F32 x2 (self + neigh),
// exact fp32 accumulation. Mean-normalization fused into the A fragment.
// ---------------------------------------------------------------------------
__global__ __launch_bounds__(256) void sage_wmma_kernel(
    const float* __restrict__ feat,
    const float* __restrict__ neigh_sum,
    const float* __restrict__ deg,
    const float* __restrict__ Wself,
    const float* __restrict__ bself,
    const float* __restrict__ Wneigh,
    const float* __restrict__ bneigh,
    const int*   __restrict__ act_flag,
    float*       __restrict__ out) {
    const int lane = threadIdx.x & 31;
    const int wv   = threadIdx.x >> 5;   // 0..7
    const int mt   = wv >> 2;            // 0..1  (row sub-tile)
    const int nt   = wv & 3;             // 0..3  (col sub-tile)
    const int row_base = blockIdx.x * 32 + mt * 16;

    const int m  = lane & 15;            // row within tile (A), col within tile (B)
    const int kh = (lane >> 4) << 1;     // K offset 0 or 2 within the k=4 slab
    const int n  = nt * 16 + m;          // global output column (0..63)

    int rowA  = row_base + m;
    int rowAc = rowA < N_NODES ? rowA : (N_NODES - 1);  // clamp loads; stores guarded

    float dg   = deg[rowAc];
    float invd = dg > 0.0f ? 1.0f / dg : 0.0f;

    const float* fRow = feat      + (size_t)rowAc * D;
    const float* hRow = neigh_sum + (size_t)rowAc * D;

    v8f acc = {};
#pragma unroll
    for (int k0 = 0; k0 < D; k0 += 4) {
        v2f a_self, a_ngh, b_self, b_ngh;
        a_self.x = fRow[k0 + kh];
        a_self.y = fRow[k0 + kh + 1];
        a_ngh.x  = hRow[k0 + kh]     * invd;
        a_ngh.y  = hRow[k0 + kh + 1] * invd;
        // B = W^T  =>  B[k][n] = W[n*D + k]
        b_self.x = Wself[(size_t)n * D + k0 + kh];
        b_self.y = Wself[(size_t)n * D + k0 + kh + 1];
        b_ngh.x  = Wneigh[(size_t)n * D + k0 + kh];
        b_ngh.y  = Wneigh[(size_t)n * D + k0 + kh + 1];

        acc = __builtin_amdgcn_wmma_f32_16x16x4_f32(
            false, a_self, false, b_self, (short)0, acc, false, false);
        acc = __builtin_amdgcn_wmma_f32_16x16x4_f32(
            false, a_ngh, false, b_ngh, (short)0, acc, false, false);
    }

    const float bias = bself[n] + bneigh[n];
    const int   act  = act_flag[0];
    const int   mofs = (lane >> 4) << 3;   // +0 or +8

#pragma unroll
    for (int v = 0; v < 8; ++v) {
        int mr = row_base + v + mofs;      // D-matrix: VGPR v -> M = v + 8*(lane>>4)
        if (mr < N_NODES) {
            float val = acc[v] + bias;
            if (act) val = val > 0.0f ? val : 0.0f;
            out[(size_t)mr * D + n] = val;
        }
    }
}

// ---------------------------------------------------------------------------
extern "C" void kernel_launch(void* const* d_in, const int* in_sizes, int n_in,
                              void* d_out, int out_size, void* d_ws, size_t ws_size,
                              hipStream_t stream) {
    (void)in_sizes; (void)n_in; (void)out_size; (void)ws_size;

    const float* feat   = (const float*)d_in[0];
    const int*   src    = (const int*)  d_in[1];
    const int*   dst    = (const int*)  d_in[2];
    const float* Wself  = (const float*)d_in[3];
    const float* bself  = (const float*)d_in[4];
    const float* Wneigh = (const float*)d_in[5];
    const float* bneigh = (const float*)d_in[6];
    const int*   act    = (const int*)  d_in[7];
    float*       out    = (float*)d_out;

    float* neigh_sum = (float*)d_ws;                       // N_NODES * D floats
    float* deg       = neigh_sum + (size_t)N_NODES * D;    // N_NODES floats

    const int zn = N_NODES * D + N_NODES;
    zero_ws_kernel<<<2048, 256, 0, stream>>>(neigh_sum, zn);

    edge_scatter_kernel<<<N_EDGES / 8, 256, 0, stream>>>(feat, src, dst,
                                                         neigh_sum, deg);

    const int gemm_blocks = (N_NODES + 31) / 32;  // 1563
    sage_wmma_kernel<<<gemm_blocks, 256, 0, stream>>>(feat, neigh_sum, deg,
                                                      Wself, bself, Wneigh,
                                                      bneigh, act, out);
}